// Rfft2d_75694503625121
// MI455X (gfx1250) — compile-verified
//
#include <hip/hip_runtime.h>

// ---------------------------------------------------------------------------
// Blockwise 8x8 rfft2 on MI455X (gfx1250) via V_WMMA_F32_16X16X4_F32.
//
// F = W X W^T, W = C + iS (8x8 real DFT cos / -sin matrices).
// Stage 1:  D1 = [C;S] (16x8) x [X1 X2] (8x16)   -> two blocks per wave
// Stage 2:  D2 = [CX;SX] (16x8) x ([C^T S^T]/64) (8x16) per block
// Re = D2[0:8,0:8] - D2[8:16,8:16],  Im = D2[0:8,8:16] + D2[8:16,0:8]
// Output (N, 65536, 8, 5, 2); 1/64 scale folded into stage-2 constants.
//
// Launch: 4096 blocks x 256 threads = 32768 waves = pairs-per-image, so each
// wave keeps one (by,bxp) position and walks the 16 images with constant
// pointer increments (uniform trip count -> no exec-mask loop overhead).
// ---------------------------------------------------------------------------

typedef float v2f __attribute__((ext_vector_type(2)));
typedef float v4f __attribute__((ext_vector_type(4)));
typedef float v8f __attribute__((ext_vector_type(8)));

// cos(2*pi*t/8) table; sin(2*pi*t/8) = COS8[(t+6)&7]
__constant__ float COS8[8] = {
    1.0f, 0.70710678118654752f, 0.0f, -0.70710678118654752f,
   -1.0f, -0.70710678118654752f, 0.0f, 0.70710678118654752f};

__device__ __forceinline__ float dftC(int m, int k) {            // cos term
  return COS8[(m * k) & 7];
}
__device__ __forceinline__ float dftS(int m, int k) {            // -sin term
  return -COS8[((m * k) + 6) & 7];
}

#define LDS_STRIDE 20                 // 16x16 tile padded to 20 floats/row
#define TILE_F     (16 * LDS_STRIDE)  // 320 floats per tile
#define WAVE_LDS   (3 * TILE_F)       // D1 tile + 2 D2 tiles per wave

#define N_IMG      16
#define IMG_FLOATS (1u << 22)         // 2048*2048
#define PAIRS_PER_IMG 32768           // 256 * 128
#define OUT_IMG_STRIDE ((size_t)PAIRS_PER_IMG * 160)

__global__ __launch_bounds__(256)
void rfft2_8x8_wmma_kernel(const float* __restrict__ x,
                           float* __restrict__ out) {
  __shared__ float lds[8 * WAVE_LDS];

  const int lane  = threadIdx.x & 31;
  const int wid   = threadIdx.x >> 5;
  const int gwave = blockIdx.x * (blockDim.x >> 5) + wid;   // 0..32767

  float* t1 = &lds[wid * WAVE_LDS];   // stage-1 result tile (stored [n][m])
  float* t2 = t1 + TILE_F;            // two stage-2 result tiles

  const int h   = lane >> 4;          // half-wave id (WMMA K-pairing)
  const int m15 = lane & 15;          // M for A-operand / N for B,C,D operands

  // Per-lane K indices for the two chained K=4 WMMAs (K = 0..7 total):
  const int k0 = 2 * h, k1 = 2 * h + 1, k2 = 4 + 2 * h, k3 = 5 + 2 * h;

  // Constant operand: Mstack[m15][k] = (m15<8) ? C[m15][k] : S[m15-8][k].
  // Same per-lane values serve as stage-1 A ([C;S]) and, scaled by 1/64,
  // as stage-2 B ([C^T S^T]) because C/S depend only on the product m*k.
  v2f a01, a23;
  a01.x = (m15 < 8) ? dftC(m15, k0) : dftS(m15 - 8, k0);
  a01.y = (m15 < 8) ? dftC(m15, k1) : dftS(m15 - 8, k1);
  a23.x = (m15 < 8) ? dftC(m15, k2) : dftS(m15 - 8, k2);
  a23.y = (m15 < 8) ? dftC(m15, k3) : dftS(m15 - 8, k3);
  const float sc = 1.0f / 64.0f;
  v2f r01 = {a01.x * sc, a01.y * sc};
  v2f r23 = {a23.x * sc, a23.y * sc};

  // ---- loop-invariant per-lane output gather pattern -------------------
  // element e = i*32+lane of the pair's 160 floats:
  //   blk=e/80, m=(e%80)/10, rv=(e%80)%10, v=rv>>1
  //   rv even: val = t[v][m]   - t[v+8][m+8]   (Re)
  //   rv odd : val = t[v+8][m] + t[v][m+8]     (Im)
  // tiles stored transposed: D2[m][nn] at t[nn*STRIDE + m]
  int   oa0[5], oa1[5];
  float osgn[5];
  #pragma unroll
  for (int i = 0; i < 5; ++i) {
    const int e   = i * 32 + lane;
    const int blk = e / 80;
    const int r   = e - blk * 80;
    const int m   = r / 10;
    const int rv  = r - m * 10;
    const int v   = rv >> 1;
    const int tb  = blk * TILE_F;
    if (rv & 1) {  // Im = t[(v+8)*S+m] + t[v*S+m+8]
      oa0[i]  = tb + v * LDS_STRIDE + m + 8;
      oa1[i]  = tb + (v + 8) * LDS_STRIDE + m;
      osgn[i] = 1.0f;
    } else {       // Re = t[v*S+m] - t[(v+8)*S+m+8]
      oa0[i]  = tb + (v + 8) * LDS_STRIDE + m + 8;
      oa1[i]  = tb + v * LDS_STRIDE + m;
      osgn[i] = -1.0f;
    }
  }

  // ---- one-time decode of this wave's (by, bxp) position ----
  // p0 = gwave in [0, 32768): by = p0>>7, bxp = p0&127
  const float* bp = x
      + (((size_t)(gwave >> 7)) << 14)          // by * 8 rows * 2048
      + (size_t)((gwave & 127) << 4)            // bxp * 16 cols
      + m15;
  float* op = out + (size_t)gwave * 160 + lane;

  for (int img = 0; img < N_IMG; ++img) {       // uniform trip count
    // prefetch next image's strip (speculative; OOB is silently dropped)
    {
      const float* bpn = bp + IMG_FLOATS;
      __builtin_prefetch(bpn + (size_t)k0 * 2048, 0, 1);
      __builtin_prefetch(bpn + (size_t)k1 * 2048, 0, 1);
      __builtin_prefetch(bpn + (size_t)k2 * 2048, 0, 1);
      __builtin_prefetch(bpn + (size_t)k3 * 2048, 0, 1);
    }

    // ---- load B operand: rows k of the 8x16 strip [X1 X2], column m15 ----
    v2f b01, b23;
    b01.x = bp[(size_t)k0 * 2048];
    b01.y = bp[(size_t)k1 * 2048];
    b23.x = bp[(size_t)k2 * 2048];
    b23.y = bp[(size_t)k3 * 2048];

    // ---- stage 1: D1 = [C;S] x [X1 X2]  (two chained K=4 WMMAs) ----
    v8f d1 = {0.f, 0.f, 0.f, 0.f, 0.f, 0.f, 0.f, 0.f};
    d1 = __builtin_amdgcn_wmma_f32_16x16x4_f32(false, a01, false, b01,
                                               (short)0, d1, false, false);
    d1 = __builtin_amdgcn_wmma_f32_16x16x4_f32(false, a23, false, b23,
                                               (short)0, d1, false, false);

    // store D1 to LDS as [n][m]: lane holds column N=m15, rows M = c + 8h
    {
      float* dst = &t1[m15 * LDS_STRIDE + 8 * h];
      v4f lo = {d1[0], d1[1], d1[2], d1[3]};
      v4f hi = {d1[4], d1[5], d1[6], d1[7]};
      *(v4f*)(dst)     = lo;
      *(v4f*)(dst + 4) = hi;
    }
    asm volatile("s_wait_dscnt 0" ::: "memory");

    // ---- stage 2: per block, D2 = P x ([C^T S^T]/64) ----
    #pragma unroll
    for (int blk = 0; blk < 2; ++blk) {
      const int base = 8 * blk;
      v2f p01, p23;                 // A operand: P[M=m15][k] = D1[m15][k+8blk]
      p01.x = t1[(base + k0) * LDS_STRIDE + m15];
      p01.y = t1[(base + k1) * LDS_STRIDE + m15];
      p23.x = t1[(base + k2) * LDS_STRIDE + m15];
      p23.y = t1[(base + k3) * LDS_STRIDE + m15];

      v8f d2 = {0.f, 0.f, 0.f, 0.f, 0.f, 0.f, 0.f, 0.f};
      d2 = __builtin_amdgcn_wmma_f32_16x16x4_f32(false, p01, false, r01,
                                                 (short)0, d2, false, false);
      d2 = __builtin_amdgcn_wmma_f32_16x16x4_f32(false, p23, false, r23,
                                                 (short)0, d2, false, false);

      float* dst = &t2[blk * TILE_F + m15 * LDS_STRIDE + 8 * h];
      v4f lo = {d2[0], d2[1], d2[2], d2[3]};
      v4f hi = {d2[4], d2[5], d2[6], d2[7]};
      *(v4f*)(dst)     = lo;
      *(v4f*)(dst + 4) = hi;
    }
    asm volatile("s_wait_dscnt 0" ::: "memory");

    // ---- combine quadrants, coalesced store of 160 floats ----
    #pragma unroll
    for (int i = 0; i < 5; ++i) {
      op[i * 32] = __builtin_fmaf(t2[oa0[i]], osgn[i], t2[oa1[i]]);
    }

    // ---- advance to next image: constant strides ----
    bp += IMG_FLOATS;
    op += OUT_IMG_STRIDE;
  }
}

extern "C" void kernel_launch(void* const* d_in, const int* in_sizes, int n_in,
                              void* d_out, int out_size, void* d_ws, size_t ws_size,
                              hipStream_t stream) {
  (void)in_sizes; (void)n_in; (void)out_size; (void)d_ws; (void)ws_size;
  const float* x = (const float*)d_in[0];
  float* out = (float*)d_out;
  // 4096 blocks * 8 waves = 32768 waves == pairs per image (required).
  dim3 grid(4096), block(256);
  hipLaunchKernelGGL(rfft2_8x8_wmma_kernel, grid, block, 0, stream, x, out);
}